// TokenReorderer_377957122268
// MI455X (gfx1250) — compile-verified
//
#include <hip/hip_runtime.h>
#include <hip/hip_bf16.h>
#include <stdint.h>

#define NUM_EXPERTS 64
#define TOP_K 8
#define N_TOKENS 131072
#define NK (N_TOKENS * TOP_K)        // 1048576 flat (token, k) slots
#define TPB 256
#define CHUNK 1024                   // flat elements per block
#define NBLOCKS (NK / CHUNK)         // 1024
#define ROUNDS (CHUNK / TPB)         // 4
#define WAVES_PER_BLOCK (TPB / 32)   // 8 (wave32)

#ifndef __has_builtin
#define __has_builtin(x) 0
#endif

// ---- gfx1250 wave32 ballot -------------------------------------------------
#if __has_builtin(__builtin_amdgcn_ballot_w32)
#define WAVE_BALLOT(p) __builtin_amdgcn_ballot_w32(p)
#else
#define WAVE_BALLOT(p) ((unsigned)__ballot(p))
#endif

// ---- gfx1250 async global->LDS path (ASYNCcnt tracked) ---------------------
// Probe round 1 showed the builtin's prototype uses int* parameters:
//   error: cannot initialize a parameter of type '__device__ int *'
// i.e. (addrspace(1) int* gsrc, addrspace(3) int* lds_dst, imm off, imm cpol).
#if __has_builtin(__builtin_amdgcn_global_load_async_to_lds_b32) && \
    __has_builtin(__builtin_amdgcn_s_wait_asynccnt)
#define HAVE_ASYNC_LDS 1
typedef __attribute__((address_space(1))) int g_i32;
typedef __attribute__((address_space(3))) int l_i32;
#define AS1I(p) ((g_i32*)(p))
#define AS3I(p) ((l_i32*)(p))
#endif

// ============================================================================
// Phase 1: per-block expert histogram (counts only; order-free so LDS atomics
// are fine). hist[b*64 + e] = #elements with expert e in flat chunk b.
// ============================================================================
__global__ void moe_hist_kernel(const int* __restrict__ experts,
                                unsigned* __restrict__ hist) {
    __shared__ unsigned sh[NUM_EXPERTS];
    const int t = threadIdx.x;
    const int b = blockIdx.x;
    if (t < NUM_EXPERTS) sh[t] = 0u;
    __syncthreads();

    // One int4 per thread covers the whole 1024-element chunk, coalesced b128.
    const int4* e4 = (const int4*)experts;
    int4 v = e4[b * (CHUNK / 4) + t];
    atomicAdd(&sh[((unsigned)v.x) & 63u], 1u);
    atomicAdd(&sh[((unsigned)v.y) & 63u], 1u);
    atomicAdd(&sh[((unsigned)v.z) & 63u], 1u);
    atomicAdd(&sh[((unsigned)v.w) & 63u], 1u);

    __syncthreads();
    if (t < NUM_EXPERTS) hist[b * NUM_EXPERTS + t] = sh[t];
}

// ============================================================================
// Phase 2: one small block. Thread e serially scans its expert column over all
// 1024 blocks (in place: hist[b][e] becomes the exclusive prefix), producing
// per-expert totals; thread 0 then builds expertBase via a 64-entry exclusive
// scan. Totals are also written to d_out (num_tokens_per_expert, as float).
// Total traffic is ~0.5 MB — negligible at 23.3 TB/s.
// ============================================================================
__global__ void moe_scan_kernel(unsigned* __restrict__ hist,
                                unsigned* __restrict__ expertBase,
                                float* __restrict__ outCounts) {
    __shared__ unsigned totals[NUM_EXPERTS];
    const int t = threadIdx.x;
    if (t < NUM_EXPERTS) {
        unsigned run = 0;
#pragma unroll 8
        for (int b = 0; b < NBLOCKS; ++b) {
            unsigned v = hist[b * NUM_EXPERTS + t];
            hist[b * NUM_EXPERTS + t] = run;   // exclusive prefix over blocks
            run += v;
        }
        totals[t] = run;
        outCounts[t] = (float)run;             // exact: run < 2^24
    }
    __syncthreads();
    if (t == 0) {
        unsigned base = 0;
#pragma unroll
        for (int e = 0; e < NUM_EXPERTS; ++e) {
            expertBase[e] = base;
            base += totals[e];
        }
    }
}

// ============================================================================
// Phase 3: stable scatter. Block b re-reads its chunk (staged into LDS via the
// gfx1250 async global->LDS path), computes each element's stable rank within
// the block using wave32 ballots + LDS cross-wave/cross-round accumulation,
// and writes score + token index to the final sorted position.
// ============================================================================
__global__ void moe_scatter_kernel(const float* __restrict__ scores,
                                   const int* __restrict__ experts,
                                   const unsigned* __restrict__ blockOff,
                                   const unsigned* __restrict__ expertBase,
                                   float* __restrict__ outScores,
                                   float* __restrict__ outIdx) {
    __shared__ unsigned s_expert[CHUNK];                       // 4 KB
    __shared__ float    s_score[CHUNK];                        // 4 KB
    __shared__ unsigned runCount[NUM_EXPERTS];                 // 256 B
    __shared__ unsigned waveHist[WAVES_PER_BLOCK][NUM_EXPERTS];// 2 KB

    const int t = threadIdx.x;
    const int b = blockIdx.x;
    const int base = b * CHUNK;

    // ---- stage chunk into LDS via ASYNCcnt-tracked async copies ----------
#ifdef HAVE_ASYNC_LDS
#pragma unroll
    for (int r = 0; r < ROUNDS; ++r) {
        const int j = r * TPB + t;
        __builtin_amdgcn_global_load_async_to_lds_b32(
            AS1I(experts + base + j), AS3I(&s_expert[j]), 0, 0);
        __builtin_amdgcn_global_load_async_to_lds_b32(
            AS1I(scores + base + j), AS3I(&s_score[j]), 0, 0);
    }
    __builtin_amdgcn_s_wait_asynccnt(0);
#else
#pragma unroll
    for (int r = 0; r < ROUNDS; ++r) {
        const int j = r * TPB + t;
        s_expert[j] = (unsigned)experts[base + j];
        s_score[j]  = scores[base + j];
    }
#endif
    if (t < NUM_EXPERTS) runCount[t] = 0u;
    __syncthreads();

    const int wave = t >> 5;
    const int lane = t & 31;
    const unsigned ltmask = lane ? (0xffffffffu >> (32 - lane)) : 0u;

    for (int r = 0; r < ROUNDS; ++r) {
        const int j = r * TPB + t;             // thread order == flat order
        const unsigned e = s_expert[j] & 63u;
        const float sc = s_score[j];

        // zero this round's cross-wave histogram (512 entries / 256 threads)
        ((unsigned*)waveHist)[t] = 0u;
        ((unsigned*)waveHist)[t + TPB] = 0u;
        __syncthreads();

        // wave-level stable rank via uniform ballot sweep over 64 experts
        unsigned same = 0u;
        for (unsigned q = 0; q < NUM_EXPERTS; ++q) {
            unsigned bal = WAVE_BALLOT(e == q);
            if (e == q) same = bal;
        }
        const unsigned rankInWave = (unsigned)__popc(same & ltmask);
        if (rankInWave == 0u)                      // per-expert wave leader
            waveHist[wave][e] = (unsigned)__popc(same);
        __syncthreads();

        // add counts from earlier waves (this round) + earlier rounds
        unsigned prefix = runCount[e];
        for (int w = 0; w < wave; ++w) prefix += waveHist[w][e];
        const unsigned rank = prefix + rankInWave;

        const unsigned dest = expertBase[e] + blockOff[b * NUM_EXPERTS + e] + rank;
        const unsigned flat = (unsigned)(base + j);
        outScores[dest] = sc;
        outIdx[dest]    = (float)(flat >> 3);      // flat / TOP_K, exact in f32
        __syncthreads();

        // fold this round's counts into the running per-expert counters
        if (t < NUM_EXPERTS) {
            unsigned s = 0u;
#pragma unroll
            for (int w = 0; w < WAVES_PER_BLOCK; ++w) s += waveHist[w][t];
            runCount[t] += s;
        }
        __syncthreads();
    }
}

// ============================================================================
extern "C" void kernel_launch(void* const* d_in, const int* in_sizes, int n_in,
                              void* d_out, int out_size, void* d_ws, size_t ws_size,
                              hipStream_t stream) {
    const float* scores  = (const float*)d_in[0];   // (N_TOKENS, TOP_K) f32
    const int*   experts = (const int*)d_in[1];     // (N_TOKENS, TOP_K) int

    float* outScores = (float*)d_out;               // NK sorted scores
    float* outIdx    = outScores + NK;              // NK sorted token indices
    float* outCounts = outScores + 2 * NK;          // 64 expert counts

    unsigned* hist       = (unsigned*)d_ws;                 // NBLOCKS*64 u32 (becomes blockOff in place)
    unsigned* expertBase = hist + NBLOCKS * NUM_EXPERTS;    // 64 u32

    moe_hist_kernel<<<NBLOCKS, TPB, 0, stream>>>(experts, hist);
    moe_scan_kernel<<<1, 64, 0, stream>>>(hist, expertBase, outCounts);
    moe_scatter_kernel<<<NBLOCKS, TPB, 0, stream>>>(scores, experts, hist,
                                                    expertBase, outScores, outIdx);
}